// HybridModel_28209345200317
// MI455X (gfx1250) — compile-verified
//
#include <hip/hip_runtime.h>

// Problem sizes (fixed by the reference)
#define NN   32768   // nodes
#define EE   131072  // edges
#define DD   128     // hidden dim
#define FF   64      // input feature dim
#define GG   128     // graphs
#define PP   256     // nodes per graph
#define CC   4       // centroids
#define RR   2       // repeats (S*KENS)
#define EPG  1024    // edges per graph
#define OUTD 10

typedef float v2f __attribute__((ext_vector_type(2)));
typedef float v8f __attribute__((ext_vector_type(8)));

// ---------------------------------------------------------------------------
// CDNA5 async global->LDS copy (ASYNCcnt-tracked DMA, no VGPR round trip).
// Flat pointers to LDS carry the LDS byte offset in their low 32 bits
// (aperture check uses addr[63:32]).
// ---------------------------------------------------------------------------
__device__ __forceinline__ void async_ld128(void* lds_dst, const void* gsrc) {
  unsigned l = (unsigned)(unsigned long long)lds_dst;
  unsigned long long ga = (unsigned long long)gsrc;
  asm volatile("global_load_async_to_lds_b128 %0, %1, off"
               :: "v"(l), "v"(ga) : "memory");
}
__device__ __forceinline__ void async_wait0() {
  asm volatile("s_wait_asynccnt 0x0" ::: "memory");
}

// ---------------------------------------------------------------------------
// Fused WMMA f32 GEMM: out[M,128] = act(alpha*(A0@W0 + A1@W1 + A2@W2)
//                                       + bias + addmat), optional +=.
// One wave computes a 16x16 tile via V_WMMA_F32_16X16X4_F32, K in steps of 4.
// A strips staged in LDS via async DMA; W read through WGP$/L2 (shared by all
// blocks). Block = 256 threads = 8 waves = the 8 column tiles of 128.
// ---------------------------------------------------------------------------
__device__ inline v8f wmma_pair(v8f acc, const float* __restrict__ lA,
                                const float* __restrict__ W, int K,
                                int col0, int m, int sub) {
  for (int k = 0; k < K; k += 4) {
    v2f a, b;
    // A 16x4 f32 layout: VGPR0 = K (lanes0-15) / K+2 (lanes16-31), VGPR1 = +1
    a.x = lA[m * K + k + 2 * sub];
    a.y = lA[m * K + k + 2 * sub + 1];
    // B 4x16 layout: VGPR0 rows {k, k+2}, VGPR1 rows {k+1, k+3}, cols by lane
    b.x = W[(size_t)(k + 2 * sub) * DD + col0 + m];
    b.y = W[(size_t)(k + 2 * sub + 1) * DD + col0 + m];
    acc = __builtin_amdgcn_wmma_f32_16x16x4_f32(false, a, false, b,
                                                (short)0, acc, false, false);
  }
  return acc;
}

__global__ __launch_bounds__(256) void gemm_wmma(
    const float* A0, const float* W0, const float* A1, const float* W1,
    const float* A2, const float* W2, const float* bias, const float* addmat,
    float* out, int M, int K, float alpha, int relu, int accum) {
  __shared__ alignas(16) float lA[3][16 * DD];
  const int tid  = threadIdx.x;
  const int row0 = blockIdx.x * 16;
  const float* As[3] = {A0, A1, A2};
  const float* Ws[3] = {W0, W1, W2};
  for (int p = 0; p < 3; ++p) {
    if (!As[p]) continue;
    // prefetch weight matrix into cache hierarchy (global_prefetch_b8)
    for (int i = tid; i < (K * DD) / 16; i += 256)
      __builtin_prefetch(Ws[p] + (size_t)i * 16, 0, 0);
    // async DMA the 16xK A strip into LDS (16B per lane, all lanes active)
    const float* Ap = As[p];
    const int n16 = (16 * K) / 4;
    for (int i = tid; i < n16; i += 256) {
      int r = (4 * i) / K, c = 4 * i - r * K;
      async_ld128(&lA[p][4 * i], Ap + (size_t)(row0 + r) * K + c);
    }
  }
  async_wait0();
  __syncthreads();
  const int lane = tid & 31, wave = tid >> 5;
  const int m = lane & 15, sub = lane >> 4;
  const int col0 = wave * 16;
  v8f acc = {};
  for (int p = 0; p < 3; ++p)
    if (As[p]) acc = wmma_pair(acc, lA[p], Ws[p], K, col0, m, sub);
  // C/D layout: VGPR v -> row row0+v (lanes0-15) / row0+v+8 (lanes16-31)
#pragma unroll
  for (int v = 0; v < 8; ++v) {
    int row = row0 + v + 8 * sub;
    int col = col0 + m;
    float val = alpha * acc[v];
    if (bias)   val += bias[col];
    if (addmat) val += addmat[(size_t)row * DD + col];
    if (relu)   val = fmaxf(val, 0.0f);
    size_t oi = (size_t)row * DD + col;
    if (accum) out[oi] += val; else out[oi] = val;
  }
}

// ---------------------------------------------------------------------------
// Scores + softmax over centroids -> nm[r=k][c][n] = softmax_c(x@W_score)[n,c,k]
// ---------------------------------------------------------------------------
__global__ __launch_bounds__(256) void score_softmax(const float* __restrict__ x,
                                                     const float* __restrict__ Wsc,
                                                     float* __restrict__ nm) {
  int tid = blockIdx.x * 256 + threadIdx.x;  // over N*KENS
  if (tid >= NN * 2) return;
  int n = tid >> 1, k = tid & 1;
  const float* xr = x + (size_t)n * DD;
  float s[CC] = {0.f, 0.f, 0.f, 0.f};
  for (int d = 0; d < DD; ++d) {
    float xv = xr[d];
#pragma unroll
    for (int c = 0; c < CC; ++c) s[c] += xv * Wsc[d * (CC * 2) + c * 2 + k];
  }
  float mx = fmaxf(fmaxf(s[0], s[1]), fmaxf(s[2], s[3]));
  float e[CC], sum = 0.f;
#pragma unroll
  for (int c = 0; c < CC; ++c) { e[c] = __expf(s[c] - mx); sum += e[c]; }
  float inv = 1.0f / sum;
#pragma unroll
  for (int c = 0; c < CC; ++c) nm[(size_t)(k * CC + c) * NN + n] = e[c] * inv;
}

// packed centroid-pair tables: triu(4,1) pairs then diagonal, 4-bit nibbles
#define PI_BITS 0x3210211000ULL
#define PJ_BITS 0x3210332321ULL

// ---------------------------------------------------------------------------
// Edge kernel: one block per graph (edges contiguous per graph).
//  - recomputes edge_emb = edge_attr @ W_edge (W_edge staged in LDS via async)
//  - num[r,c,g,:]  += sum_e nm[src]*nm[dst]^2 * (h[src]+edge_emb)   (folded agg)
//  - agg_bb[dst,:] += x[src] + edge_emb                              (atomics)
//  - Wcc[r,g,i,j] from pair weights; global max via bitwise atomicMax
// Wave handles one edge at a time; lane owns 4 of the 128 channels; per-edge
// scalars broadcast with wave shuffles (ds_bpermute).
// ---------------------------------------------------------------------------
__global__ __launch_bounds__(256) void edge_kernel(
    const int* __restrict__ src, const int* __restrict__ dst,
    const float* __restrict__ edge_attr, const float* __restrict__ Wedge,
    const float* __restrict__ nm, const float* __restrict__ h,
    const float* __restrict__ x, float* __restrict__ agg_bb,
    float* __restrict__ num, float* __restrict__ Wcc,
    unsigned int* __restrict__ wmax) {
  __shared__ alignas(16) float sW[16 * DD];
  __shared__ float sNum[8 * DD];
  __shared__ float sPW[32];
  const int tid = threadIdx.x, g = blockIdx.x;
  const int wave = tid >> 5, lane = tid & 31;
  for (int i = tid; i < (16 * DD) / 4; i += 256)
    async_ld128(&sW[4 * i], Wedge + (size_t)4 * i);
  for (int i = tid; i < 8 * DD; i += 256) sNum[i] = 0.0f;
  if (tid < 32) sPW[tid] = 0.0f;
  async_wait0();
  __syncthreads();

  float numreg[8][4];
#pragma unroll
  for (int rc = 0; rc < 8; ++rc)
#pragma unroll
    for (int j = 0; j < 4; ++j) numreg[rc][j] = 0.0f;
  float pw = 0.0f;
  const int pr  = lane / 10, pp = lane % 10;  // meaningful for lane<20
  const int ppi = (int)((PI_BITS >> (4 * pp)) & 0xF);
  const int ppj = (int)((PJ_BITS >> (4 * pp)) & 0xF);

  for (int i = wave; i < EPG; i += 8) {
    int e = g * EPG + i;
    int s = src[e], t = dst[e];
    // one load per lane, broadcast via shuffles:
    //  lanes 0-7: nm[rc][src], 8-15: nm[rc][dst], 16-31: edge_attr[e][0:16]
    float ld;
    if (lane < 8)       ld = nm[(size_t)lane * NN + s];
    else if (lane < 16) ld = nm[(size_t)(lane - 8) * NN + t];
    else                ld = edge_attr[(size_t)e * 16 + (lane - 16)];
    float a[8], b[8], ea[16];
#pragma unroll
    for (int q = 0; q < 8; ++q) { a[q] = __shfl(ld, q, 32); b[q] = __shfl(ld, 8 + q, 32); }
#pragma unroll
    for (int q = 0; q < 16; ++q) ea[q] = __shfl(ld, 16 + q, 32);
    // edge embedding: this lane's 4 channels
    float4 ee = {0.f, 0.f, 0.f, 0.f};
#pragma unroll
    for (int q = 0; q < 16; ++q) {
      float4 wv = ((const float4*)(sW + q * DD))[lane];
      ee.x += ea[q] * wv.x; ee.y += ea[q] * wv.y;
      ee.z += ea[q] * wv.z; ee.w += ea[q] * wv.w;
    }
    float4 hv = ((const float4*)(h + (size_t)s * DD))[lane];
    float4 xv = ((const float4*)(x + (size_t)s * DD))[lane];
    float mj[4] = {hv.x + ee.x, hv.y + ee.y, hv.z + ee.z, hv.w + ee.w};
#pragma unroll
    for (int rc = 0; rc < 8; ++rc) {
      float w3 = a[rc] * b[rc] * b[rc];  // nm[dst] * em
      numreg[rc][0] += w3 * mj[0]; numreg[rc][1] += w3 * mj[1];
      numreg[rc][2] += w3 * mj[2]; numreg[rc][3] += w3 * mj[3];
    }
    float* bb = agg_bb + (size_t)t * DD + lane * 4;
    atomicAdd(bb + 0, xv.x + ee.x); atomicAdd(bb + 1, xv.y + ee.y);
    atomicAdd(bb + 2, xv.z + ee.z); atomicAdd(bb + 3, xv.w + ee.w);
    if (lane < 20) pw += a[pr * 4 + ppi] * b[pr * 4 + ppj];
  }
#pragma unroll
  for (int rc = 0; rc < 8; ++rc)
#pragma unroll
    for (int j = 0; j < 4; ++j)
      atomicAdd(&sNum[rc * DD + lane * 4 + j], numreg[rc][j]);
  if (lane < 20) atomicAdd(&sPW[lane], pw);
  __syncthreads();
  for (int i = tid; i < 8 * DD; i += 256) {
    int rc = i >> 7, d = i & 127;  // block owns num[.,.,g,.] -> plain store
    num[(size_t)rc * (GG * DD) + (size_t)g * DD + d] = sNum[i];
  }
  if (tid < 20) {
    int r = tid / 10, p = tid % 10;
    float w = sPW[tid];
    int i4 = (int)((PI_BITS >> (4 * p)) & 0xF);
    int j4 = (int)((PJ_BITS >> (4 * p)) & 0xF);
    if (p >= 6) w *= 0.5f;  // diagonal counted once
    size_t base = ((size_t)r * GG + g) * 16;
    Wcc[base + i4 * 4 + j4] = w;
    Wcc[base + j4 * 4 + i4] = w;  // to_undirected (same cell for diagonal)
    atomicMax(wmax, __float_as_uint(w));  // w >= 0: bit order == float order
  }
}

// ---------------------------------------------------------------------------
// Per-graph node pooling: num += sum_n nm*h ; denom = sum_n nm ;
// agg_bc[r,g,c,:] = sum_n nm * x
// ---------------------------------------------------------------------------
__global__ __launch_bounds__(256) void node_pool(
    const float* __restrict__ nm, const float* __restrict__ h,
    const float* __restrict__ x, float* __restrict__ num,
    float* __restrict__ denom, float* __restrict__ agg_bc) {
  __shared__ float sNum[8 * DD];
  __shared__ float sBC[8 * DD];
  __shared__ float sDen[8];
  const int tid = threadIdx.x, g = blockIdx.x;
  const int wave = tid >> 5, lane = tid & 31;
  for (int i = tid; i < 8 * DD; i += 256) { sNum[i] = 0.f; sBC[i] = 0.f; }
  if (tid < 8) sDen[tid] = 0.f;
  __syncthreads();
  float nreg[8][4], breg[8][4], dreg[8];
#pragma unroll
  for (int rc = 0; rc < 8; ++rc) {
    dreg[rc] = 0.f;
#pragma unroll
    for (int j = 0; j < 4; ++j) { nreg[rc][j] = 0.f; breg[rc][j] = 0.f; }
  }
  for (int i = wave; i < PP; i += 8) {
    int n = g * PP + i;
    float ld = (lane < 8) ? nm[(size_t)lane * NN + n] : 0.0f;
    float a[8];
#pragma unroll
    for (int q = 0; q < 8; ++q) a[q] = __shfl(ld, q, 32);
    float4 hv = ((const float4*)(h + (size_t)n * DD))[lane];
    float4 xv = ((const float4*)(x + (size_t)n * DD))[lane];
#pragma unroll
    for (int rc = 0; rc < 8; ++rc) {
      nreg[rc][0] += a[rc] * hv.x; nreg[rc][1] += a[rc] * hv.y;
      nreg[rc][2] += a[rc] * hv.z; nreg[rc][3] += a[rc] * hv.w;
      breg[rc][0] += a[rc] * xv.x; breg[rc][1] += a[rc] * xv.y;
      breg[rc][2] += a[rc] * xv.z; breg[rc][3] += a[rc] * xv.w;
      dreg[rc] += a[rc];
    }
  }
#pragma unroll
  for (int rc = 0; rc < 8; ++rc)
#pragma unroll
    for (int j = 0; j < 4; ++j) {
      atomicAdd(&sNum[rc * DD + lane * 4 + j], nreg[rc][j]);
      atomicAdd(&sBC[rc * DD + lane * 4 + j], breg[rc][j]);
    }
  if (lane == 0)
#pragma unroll
    for (int rc = 0; rc < 8; ++rc) atomicAdd(&sDen[rc], dreg[rc]);
  __syncthreads();
  for (int i = tid; i < 8 * DD; i += 256) {
    int rc = i >> 7, d = i & 127, r = rc >> 2, c = rc & 3;
    num[(size_t)rc * (GG * DD) + (size_t)g * DD + d] += sNum[i];
    agg_bc[(((size_t)r * GG + g) * CC + c) * DD + d] = sBC[i];
  }
  if (tid < 8) denom[(size_t)tid * GG + g] = sDen[tid];
}

// cent_x[r,g,c,d] = num[r,c,g,d] / (denom[r,c,g] + 1e-6)
__global__ void cent_kernel(const float* __restrict__ num,
                            const float* __restrict__ denom,
                            float* __restrict__ cent) {
  int idx = blockIdx.x * 256 + threadIdx.x;  // R*G*C*D = 131072
  if (idx >= RR * GG * CC * DD) return;
  int d = idx & 127, c = (idx >> 7) & 3, g = (idx >> 9) & 127, r = idx >> 16;
  int rc = r * 4 + c;
  cent[idx] = num[(size_t)rc * (GG * DD) + (size_t)g * DD + d] /
              (denom[rc * GG + g] + 1e-6f);
}

// agg_cc[r,g,i,:] = (Wcc/(max+1e-9)) @ cent_x    (tiny 4x4 per graph)
__global__ void aggcc_kernel(const float* __restrict__ Wcc,
                             const unsigned int* __restrict__ wmax,
                             const float* __restrict__ cent,
                             float* __restrict__ aggcc) {
  int idx = blockIdx.x * 256 + threadIdx.x;
  if (idx >= RR * GG * CC * DD) return;
  int d = idx & 127, ci = (idx >> 7) & 3, g = (idx >> 9) & 127, r = idx >> 16;
  float inv = 1.0f / (__uint_as_float(*wmax) + 1e-9f);
  size_t base = (size_t)r * GG + g;
  float s = 0.f;
#pragma unroll
  for (int j = 0; j < CC; ++j)
    s += Wcc[base * 16 + ci * 4 + j] * cent[(base * CC + j) * DD + d];
  aggcc[idx] = s * inv;
}

// agg_cb[r,n,:] = sum_c nm[r,c,n] * h_cent[r,batch[n],c,:]
__global__ void aggcb_kernel(const float* __restrict__ nm,
                             const float* __restrict__ hcent,
                             const int* __restrict__ batch,
                             float* __restrict__ aggcb) {
  int tid = blockIdx.x * 256 + threadIdx.x;  // R*N*32
  if (tid >= RR * NN * 32) return;
  int r = tid >> 20, rem = tid & ((NN * 32) - 1);
  int n = rem >> 5, q = rem & 31;
  int g = batch[n];
  float4 res = {0.f, 0.f, 0.f, 0.f};
#pragma unroll
  for (int c = 0; c < CC; ++c) {
    float w = nm[(size_t)(r * CC + c) * NN + n];
    float4 hc = ((const float4*)(hcent + ((size_t)(r * GG + g) * CC + c) * DD))[q];
    res.x += w * hc.x; res.y += w * hc.y; res.z += w * hc.z; res.w += w * hc.w;
  }
  ((float4*)(aggcb + ((size_t)r * NN + n) * DD))[q] = res;
}

// graph mean pool + output head
__global__ __launch_bounds__(256) void final_kernel(
    const float* __restrict__ node_emb, const float* __restrict__ Wout,
    const float* __restrict__ bout, float* __restrict__ out) {
  __shared__ float sGE[DD];
  const int tid = threadIdx.x, g = blockIdx.x;
  const int d = tid & 127, half = tid >> 7;
  float acc = 0.f;
  for (int i = half; i < PP; i += 2)
    acc += node_emb[((size_t)g * PP + i) * DD + d];
  if (half == 0) sGE[d] = acc;
  __syncthreads();
  if (half == 1) atomicAdd(&sGE[d], acc);
  __syncthreads();
  if (tid < OUTD) {
    float v = bout[tid];
    for (int dd = 0; dd < DD; ++dd)
      v += (sGE[dd] * (1.0f / PP)) * Wout[dd * OUTD + tid];
    out[g * OUTD + tid] = v;
  }
}

__global__ void zero_kernel(float* __restrict__ p, int n) {
  for (int i = blockIdx.x * blockDim.x + threadIdx.x; i < n;
       i += gridDim.x * blockDim.x)
    p[i] = 0.0f;
}

// ---------------------------------------------------------------------------
extern "C" void kernel_launch(void* const* d_in, const int* in_sizes, int n_in,
                              void* d_out, int out_size, void* d_ws, size_t ws_size,
                              hipStream_t stream) {
  (void)in_sizes; (void)n_in; (void)out_size; (void)ws_size;
  const float* x_feat    = (const float*)d_in[0];
  const float* edge_attr = (const float*)d_in[1];
  const float* W_atom    = (const float*)d_in[2];
  const float* b_atom    = (const float*)d_in[3];
  const float* W_score   = (const float*)d_in[4];
  const float* W_edge    = (const float*)d_in[5];
  const float* W_b2c     = (const float*)d_in[6];
  const float* W_bb      = (const float*)d_in[7];
  const float* W_bc      = (const float*)d_in[8];
  const float* W_cb      = (const float*)d_in[9];
  const float* W_cc      = (const float*)d_in[10];
  const float* W_sb      = (const float*)d_in[11];
  const float* W_sc      = (const float*)d_in[12];
  const float* W_inter   = (const float*)d_in[13];
  const float* b_inter   = (const float*)d_in[14];
  const float* W_out     = (const float*)d_in[15];
  const float* b_out     = (const float*)d_in[16];
  const int*   edge_idx  = (const int*)d_in[17];
  const int*   batch     = (const int*)d_in[18];
  const int* src = edge_idx;
  const int* dst = edge_idx + EE;

  float* ws = (float*)d_ws;
  size_t o = 0;
  auto take = [&](size_t n) { float* p = ws + o; o += n; return p; };
  float* X       = take((size_t)NN * DD);
  float* H       = take((size_t)NN * DD);
  float* NM      = take((size_t)RR * CC * NN);
  float* AGGBB   = take((size_t)NN * DD);
  float* NUM     = take((size_t)RR * CC * GG * DD);
  float* DEN     = take((size_t)RR * CC * GG);
  float* AGGBC   = take((size_t)RR * CC * GG * DD);
  float* CENT    = take((size_t)RR * CC * GG * DD);
  float* WCC     = take((size_t)RR * GG * CC * CC);
  float* WMAXF   = take(16);
  float* AGGCC   = take((size_t)RR * CC * GG * DD);
  float* HCENT   = take((size_t)RR * CC * GG * DD);
  float* AGGCB   = take((size_t)RR * NN * DD);
  float* B0      = take((size_t)NN * DD);
  float* HBSUM   = take((size_t)NN * DD);
  float* NODEEMB = take((size_t)NN * DD);

  zero_kernel<<<1024, 256, 0, stream>>>(AGGBB, NN * DD);
  zero_kernel<<<1, 64, 0, stream>>>(WMAXF, 16);
  // x = x_feat @ W_atom + b_atom
  gemm_wmma<<<NN / 16, 256, 0, stream>>>(x_feat, W_atom, nullptr, nullptr,
      nullptr, nullptr, b_atom, nullptr, X, NN, FF, 1.0f, 0, 0);
  // nm = softmax over centroids of (x @ W_score)
  score_softmax<<<(NN * 2) / 256, 256, 0, stream>>>(X, W_score, NM);
  // h = relu(x @ W_b2c)
  gemm_wmma<<<NN / 16, 256, 0, stream>>>(X, W_b2c, nullptr, nullptr,
      nullptr, nullptr, nullptr, nullptr, H, NN, DD, 1.0f, 1, 0);
  // edge pass: num (edge part), agg_bb, Wcc, global max
  edge_kernel<<<GG, 256, 0, stream>>>(src, dst, edge_attr, W_edge, NM, H, X,
      AGGBB, NUM, WCC, (unsigned int*)WMAXF);
  // node pass: num += nm*h ; denom ; agg_bc
  node_pool<<<GG, 256, 0, stream>>>(NM, H, X, NUM, DEN, AGGBC);
  cent_kernel<<<(RR * CC * GG * DD) / 256, 256, 0, stream>>>(NUM, DEN, CENT);
  aggcc_kernel<<<(RR * CC * GG * DD) / 256, 256, 0, stream>>>(
      WCC, (const unsigned int*)WMAXF, CENT, AGGCC);
  // h_cent = relu(agg_bc@W_bc + agg_cc@W_cc + cent_x@W_sc)  [1024 x 128]
  gemm_wmma<<<(RR * GG * CC) / 16, 256, 0, stream>>>(AGGBC, W_bc, AGGCC, W_cc,
      CENT, W_sc, nullptr, nullptr, HCENT, RR * GG * CC, DD, 1.0f, 1, 0);
  aggcb_kernel<<<(RR * NN * 32) / 256, 256, 0, stream>>>(NM, HCENT, batch, AGGCB);
  // B0 = agg_bb@W_bb + x@W_sb (r-independent part of h_base)
  gemm_wmma<<<NN / 16, 256, 0, stream>>>(AGGBB, W_bb, X, W_sb, nullptr, nullptr,
      nullptr, nullptr, B0, NN, DD, 1.0f, 0, 0);
  // hb_sum = relu(agg_cb_0@W_cb + B0) + relu(agg_cb_1@W_cb + B0)
  gemm_wmma<<<NN / 16, 256, 0, stream>>>(AGGCB, W_cb, nullptr, nullptr,
      nullptr, nullptr, nullptr, B0, HBSUM, NN, DD, 1.0f, 1, 0);
  gemm_wmma<<<NN / 16, 256, 0, stream>>>(AGGCB + (size_t)NN * DD, W_cb, nullptr,
      nullptr, nullptr, nullptr, nullptr, B0, HBSUM, NN, DD, 1.0f, 1, 1);
  // node_emb = mean_r(h_base)@W_inter + b_inter  (mean folded into alpha=0.5)
  gemm_wmma<<<NN / 16, 256, 0, stream>>>(HBSUM, W_inter, nullptr, nullptr,
      nullptr, nullptr, b_inter, nullptr, NODEEMB, NN, DD, 0.5f, 0, 0);
  // graph mean pool + output head
  final_kernel<<<GG, 256, 0, stream>>>(NODEEMB, W_out, b_out, (float*)d_out);
}